// GATLayerAdj_89154931130992
// MI455X (gfx1250) — compile-verified
//
#include <hip/hip_runtime.h>
#include <hip/hip_bf16.h>
#include <math.h>

typedef __attribute__((ext_vector_type(2))) float v2f;
typedef __attribute__((ext_vector_type(8))) float v8f;

#define N_NODES 512
#define D_FEAT  128

// ---------------------------------------------------------------------------
// Kernel A: f_src = x @ Wf[:, :128]^T and f_tgt = x @ Wf[:, 128:]^T
// via V_WMMA_F32_16X16X4_F32. One 16x16 output tile per wave.
// Tile space: 2 matrices x (512/16=32 M-tiles) x (128/16=8 N-tiles) = 512 tiles.
// 4 waves per 128-thread block -> 128 blocks.
// ---------------------------------------------------------------------------
__global__ void __launch_bounds__(128)
gat_fgemm_wmma(const float* __restrict__ x,   // (512,128) row-major
               const float* __restrict__ Wf,  // (128,256) row-major
               float* __restrict__ fsrc,      // (512,128)
               float* __restrict__ ftgt)      // (512,128)
{
    const int lane = threadIdx.x & 31;
    const int wid  = blockIdx.x * (blockDim.x >> 5) + (threadIdx.x >> 5);

    const int matrix = wid >> 8;      // 0 = src half of Wf, 1 = tgt half
    const int rem    = wid & 255;
    const int tm     = rem >> 3;      // M tile 0..31
    const int tn     = rem & 7;       // N tile 0..7

    const int half = lane >> 4;       // 0: lanes 0-15, 1: lanes 16-31
    const int l15  = lane & 15;
    const int woff = matrix * D_FEAT; // column offset inside Wf row (256 wide)

    // A-matrix 16x4 f32 layout: M = lane&15; VGPR0 holds K=2*half, VGPR1 K=2*half+1
    // B-matrix 4x16 f32 layout: N = lane&15; VGPR0 holds K=2*half, VGPR1 K=2*half+1
    const int m = tm * 16 + l15;      // row of x this lane supplies
    const int d = tn * 16 + l15;      // row of Wf (= output column) this lane supplies

    v8f c = {};
    const float* xrow = x + m * D_FEAT;
    const float* wrow = Wf + d * 256 + woff;

    #pragma unroll 4
    for (int kk = 0; kk < D_FEAT; kk += 4) {
        const int ka = kk + 2 * half;
        v2f a, b;
        a.x = xrow[ka + 0];
        a.y = xrow[ka + 1];
        b.x = wrow[ka + 0];
        b.y = wrow[ka + 1];
        // D = A(16x4) * B(4x16) + C, fp32 in / fp32 accumulate
        c = __builtin_amdgcn_wmma_f32_16x16x4_f32(
                /*neg_a=*/false, a, /*neg_b=*/false, b,
                /*c_mod=*/(short)0, c, /*reuse_a=*/false, /*reuse_b=*/false);
    }

    // C/D 16x16 f32 layout: VGPR v -> M = v + 8*half, N = lane&15
    float* __restrict__ F = matrix ? ftgt : fsrc;
    const int ncol = tn * 16 + l15;
    #pragma unroll
    for (int v = 0; v < 8; ++v) {
        const int row = tm * 16 + v + 8 * half;
        F[row * D_FEAT + ncol] = c[v];
    }
}

// ---------------------------------------------------------------------------
// Kernel B: s[j] = x[j,:] . Ww[0,:128],  t[j] = x[j,:] . Ww[0,128:]
// One thread per row j. 4 blocks x 128 threads.
// ---------------------------------------------------------------------------
__global__ void __launch_bounds__(128)
gat_st(const float* __restrict__ x,
       const float* __restrict__ Ww,   // (1,256)
       float* __restrict__ s,
       float* __restrict__ t)
{
    const int j = blockIdx.x * blockDim.x + threadIdx.x;
    if (j >= N_NODES) return;
    const float* xrow = x + j * D_FEAT;
    float acc_s = 0.f, acc_t = 0.f;
    #pragma unroll 8
    for (int k = 0; k < D_FEAT; ++k) {
        const float xv = xrow[k];
        acc_s = fmaf(xv, Ww[k],          acc_s);
        acc_t = fmaf(xv, Ww[D_FEAT + k], acc_t);
    }
    s[j] = acc_s;
    t[j] = acc_t;
}

// ---------------------------------------------------------------------------
// Kernel C: per-row masked softmax + relu-weighted aggregation.
// One block (128 threads = 4 wave32) per node i.
//   a[i,j]    = adj[i,j] > 0 ? s[j] + t[i] + bw : -1e16
//   att[i,:]  = softmax(a[i,:])
//   o[i,d]    = sum_j att[i,j] * relu(fsrc[j,d] + ftgt[i,d] + bf[d])
// ---------------------------------------------------------------------------
__global__ void __launch_bounds__(128)
gat_row(const int*   __restrict__ adj,   // (512,512)
        const float* __restrict__ bf,    // (128,)
        const float* __restrict__ bw,    // (1,)
        const float* __restrict__ s,     // (512,)
        const float* __restrict__ t,     // (512,)
        const float* __restrict__ fsrc,  // (512,128)
        const float* __restrict__ ftgt,  // (512,128)
        float* __restrict__ out_o,       // (512,128)
        float* __restrict__ out_att)     // (512,512)
{
    const int i   = blockIdx.x;
    const int tid = threadIdx.x;  // 0..127

    __shared__ float att[N_NODES];
    __shared__ float sred[128];

    const float ti  = t[i];
    const float bw0 = bw[0];

    // ---- masked scores (each thread owns 4 j's) + row max ----
    float loc[4];
    float m = -INFINITY;
    #pragma unroll
    for (int q = 0; q < 4; ++q) {
        const int j = tid + q * 128;
        const float v = (adj[i * N_NODES + j] > 0) ? (s[j] + ti + bw0) : -1e16f;
        loc[q] = v;
        m = fmaxf(m, v);
    }
    sred[tid] = m;
    __syncthreads();
    for (int off = 64; off > 0; off >>= 1) {
        if (tid < off) sred[tid] = fmaxf(sred[tid], sred[tid + off]);
        __syncthreads();
    }
    const float rowmax = sred[0];
    __syncthreads();

    // ---- exp + row sum ----
    float psum = 0.f;
    #pragma unroll
    for (int q = 0; q < 4; ++q) {
        const float e = __expf(loc[q] - rowmax);
        att[tid + q * 128] = e;
        psum += e;
    }
    sred[tid] = psum;
    __syncthreads();
    for (int off = 64; off > 0; off >>= 1) {
        if (tid < off) sred[tid] += sred[tid + off];
        __syncthreads();
    }
    const float inv = 1.f / sred[0];
    __syncthreads();

    // ---- normalize, emit a_att ----
    #pragma unroll
    for (int q = 0; q < 4; ++q) {
        const int j = tid + q * 128;
        const float a = att[j] * inv;
        att[j] = a;
        out_att[i * N_NODES + j] = a;
    }
    __syncthreads();

    // ---- o[i,d] : thread owns d = tid ----
    const float ftd = ftgt[i * D_FEAT + tid] + bf[tid];
    float acc = 0.f;
    #pragma unroll 4
    for (int j = 0; j < N_NODES; ++j) {
        const float y = fmaxf(fsrc[j * D_FEAT + tid] + ftd, 0.f);
        acc = fmaf(att[j], y, acc);
    }
    out_o[i * D_FEAT + tid] = acc;
}

// ---------------------------------------------------------------------------
// Host entry. Inputs: x, adj, Wf, bf, Ww, bw. Output: concat(o, a_att).
// Workspace layout (floats): s[512] | t[512] | fsrc[512*128] | ftgt[512*128]
// ---------------------------------------------------------------------------
extern "C" void kernel_launch(void* const* d_in, const int* in_sizes, int n_in,
                              void* d_out, int out_size, void* d_ws, size_t ws_size,
                              hipStream_t stream) {
    const float* x   = (const float*)d_in[0];
    const int*   adj = (const int*)  d_in[1];
    const float* Wf  = (const float*)d_in[2];
    const float* bf  = (const float*)d_in[3];
    const float* Ww  = (const float*)d_in[4];
    const float* bw  = (const float*)d_in[5];

    float* ws   = (float*)d_ws;
    float* s    = ws;
    float* t    = ws + N_NODES;
    float* fsrc = ws + 2 * N_NODES;
    float* ftgt = fsrc + N_NODES * D_FEAT;

    float* out_o   = (float*)d_out;                    // (512,128)
    float* out_att = out_o + N_NODES * D_FEAT;         // (512,512)

    // A: WMMA GEMMs for f_src / f_tgt. 512 tiles, 4 waves/block -> 128 blocks.
    gat_fgemm_wmma<<<128, 128, 0, stream>>>(x, Wf, fsrc, ftgt);
    // B: attention matvecs.
    gat_st<<<4, 128, 0, stream>>>(x, Ww, s, t);
    // C: softmax + aggregation, one block per node.
    gat_row<<<N_NODES, 128, 0, stream>>>(adj, bf, bw, s, t, fsrc, ftgt, out_o, out_att);
}